// GCNsimple_64982855188731
// MI455X (gfx1250) — compile-verified
//
#include <hip/hip_runtime.h>
#include <hip/hip_bf16.h>

typedef __attribute__((ext_vector_type(2))) float v2f;
typedef __attribute__((ext_vector_type(8))) float v8f;
typedef __attribute__((ext_vector_type(4))) unsigned v4u;
typedef __attribute__((ext_vector_type(8))) unsigned v8u;

// ---------------------------------------------------------------------------
// Tensor Data Mover: 2D tile load Global -> LDS (D# per CDNA5 ISA)
// 2-group form (VADDR2/3 = NULL): tensors up to 2D. All operands uniform.
// OOB rows (y >= tensor_d1) are zero-filled by hardware -> free tail handling.
// ---------------------------------------------------------------------------
__device__ __forceinline__ void tdm_load_2d(unsigned lds_off, const void* gptr,
                                            unsigned tensor_d0, unsigned tensor_d1,
                                            unsigned tile_d0, unsigned tile_d1,
                                            unsigned stride0_elems) {
    unsigned long long ga = (unsigned long long)gptr;
    v4u g0;
    g0[0] = 1u;                                             // count=1 (valid user D#)
    g0[1] = lds_off;                                        // lds_addr (bytes)
    g0[2] = (unsigned)ga;                                   // global_addr[31:0]
    g0[3] = ((unsigned)(ga >> 32) & 0x01FFFFFFu) | (2u << 30); // addr[56:32] | type=2
    v8u g1;
    g1[0] = 2u << 16;                                       // data_size = 4 bytes
    g1[1] = (tensor_d0 & 0xFFFFu) << 16;                    // tensor_dim0[15:0]
    g1[2] = (tensor_d0 >> 16) | ((tensor_d1 & 0xFFFFu) << 16); // td0[31:16] | td1[15:0]
    g1[3] = (tensor_d1 >> 16) | (tile_d0 << 16);            // td1[31:16] | tile_dim0
    g1[4] = tile_d1;                                        // tile_dim1 (tile_dim2=0)
    g1[5] = stride0_elems;                                  // tensor_dim0_stride[31:0]
    g1[6] = 0u;                                             // stride0[47:32]=0, stride1 lo=0
    g1[7] = 0u;                                             // stride1 hi=0
    asm volatile("tensor_load_to_lds %0, %1" :: "s"(g0), "s"(g1) : "memory");
}

// ---------------------------------------------------------------------------
// Small utility kernels
// ---------------------------------------------------------------------------
__global__ __launch_bounds__(256) void fill_zero_kernel(float* __restrict__ p, int n) {
    int i = blockIdx.x * 256 + threadIdx.x;
    if (i < n) p[i] = 0.0f;
}

__global__ __launch_bounds__(256) void bias_fill_kernel(float* __restrict__ buf,
                                                        const float* __restrict__ bias,
                                                        int D, long long total) {
    long long i = (long long)blockIdx.x * 256 + threadIdx.x;
    if (i < total) buf[i] = bias[(int)(i % D)];
}

__global__ __launch_bounds__(256) void relu_kernel(float* __restrict__ p, long long n) {
    long long i = (long long)blockIdx.x * 256 + threadIdx.x;
    if (i < n) p[i] = fmaxf(p[i], 0.0f);
}

__global__ __launch_bounds__(256) void degree_kernel(const int* __restrict__ dst,
                                                     float* __restrict__ deg, int E) {
    int e = blockIdx.x * 256 + threadIdx.x;
    if (e < E) atomicAdd(&deg[dst[e]], 1.0f);
}

__global__ __launch_bounds__(256) void inv_sqrt_kernel(float* __restrict__ deg, int n) {
    int i = blockIdx.x * 256 + threadIdx.x;
    if (i < n) {
        float d = deg[i];
        deg[i] = (d > 0.0f) ? rsqrtf(fmaxf(d, 1.0f)) : 0.0f;
    }
}

__global__ __launch_bounds__(256) void norm_kernel(const int* __restrict__ src,
                                                   const int* __restrict__ dst,
                                                   const float* __restrict__ dinv,
                                                   float* __restrict__ norm, int E) {
    int e = blockIdx.x * 256 + threadIdx.x;
    if (e < E) norm[e] = dinv[src[e]] * dinv[dst[e]];
}

// ---------------------------------------------------------------------------
// Edge aggregation: out[dst[e]] += H[src[e]] * norm[e]  (scatter-add via L2
// f32 atomics; destination matrix fits in the 192MB L2). One thread per
// (edge, float4 chunk), chunk fastest so source-row gathers are coalesced.
// ---------------------------------------------------------------------------
template <int D>
__global__ __launch_bounds__(256)
void aggregate_kernel(const float* __restrict__ H, const float* __restrict__ norm,
                      const int* __restrict__ src, const int* __restrict__ dst,
                      float* __restrict__ out, int E) {
    constexpr int C4 = D / 4;
    long long idx = (long long)blockIdx.x * 256 + threadIdx.x;
    long long total = (long long)E * C4;
    if (idx >= total) return;
    int e = (int)(idx / C4);
    int c = (int)(idx % C4);
    int s = src[e];
    int d = dst[e];
    float w = norm[e];
    const float4 v = ((const float4*)(H + (size_t)s * D))[c];
    float* o = out + (size_t)d * D + (size_t)c * 4;
    atomicAdd(o + 0, v.x * w);
    atomicAdd(o + 1, v.y * w);
    atomicAdd(o + 2, v.z * w);
    atomicAdd(o + 3, v.w * w);
}

// ---------------------------------------------------------------------------
// WMMA f32 GEMM:  Y[nrows, NCOLS] = X[nrows, 128] @ W[128, NCOLS]
// Block = 256 threads (8 waves), tile = 128 rows x NCOLS.
// X tile and W are DMA'd into LDS by the Tensor Data Mover (wave 0), then each
// wave runs a double-buffered K loop of V_WMMA_F32_16X16X4_F32 with LDS-fed
// A/B fragments so ds loads for step k+4 overlap the WMMAs of step k.
// ---------------------------------------------------------------------------
template <int NCOLS>
__global__ __launch_bounds__(256)
void gemm_wmma_kernel(const float* __restrict__ X, const float* __restrict__ W,
                      float* __restrict__ Y, int nrows) {
    constexpr int K = 128;
    constexpr int ROWS = 128;                 // rows per block (8 m-tiles)
    constexpr int NTN = NCOLS / 16;           // n-tiles
    constexpr int MT = NTN;                   // m-tiles per wave (8 waves cover all)

    __shared__ float LX[ROWS][K];             // 64 KB, TDM-filled
    __shared__ float LW[K][NCOLS];            // 64/32 KB, TDM-filled (row-major)

    const int tid = threadIdx.x;
    const int wave = tid >> 5;
    const int lane = tid & 31;
    const int row0 = blockIdx.x * ROWS;

    if (tid < 32) {  // wave 0 issues both DMAs (EXEC ignored by TDM)
        unsigned remRows = (unsigned)(nrows - row0);   // rows beyond -> zero-fill
        tdm_load_2d((unsigned)(unsigned long long)&LX[0][0],
                    X + (size_t)row0 * K,
                    /*tensor_d0=*/K, /*tensor_d1=*/remRows,
                    /*tile_d0=*/K, /*tile_d1=*/ROWS, /*stride0=*/K);
        tdm_load_2d((unsigned)(unsigned long long)&LW[0][0],
                    W,
                    /*tensor_d0=*/NCOLS, /*tensor_d1=*/K,
                    /*tile_d0=*/NCOLS, /*tile_d1=*/K, /*stride0=*/NCOLS);
        __builtin_amdgcn_s_wait_tensorcnt(0);
    }
    __syncthreads();

    const int nt    = wave % NTN;             // n-tile of this wave
    const int mt0   = (wave / NTN) * MT;      // first m-tile of this wave
    const int ln    = lane & 15;
    const int koff  = (lane < 16) ? 0 : 2;    // frag K offset per ISA 16x4 layout
    const int ncol  = nt * 16 + ln;

    const v8f vzero = {0.f, 0.f, 0.f, 0.f, 0.f, 0.f, 0.f, 0.f};
    v8f acc[MT];
#pragma unroll
    for (int m = 0; m < MT; ++m) acc[m] = vzero;

    // Prologue: fragments for k = 0.
    v2f a[MT], b;
    b.x = LW[koff][ncol];
    b.y = LW[koff + 1][ncol];
#pragma unroll
    for (int m = 0; m < MT; ++m)
        a[m] = *(const v2f*)(&LX[(mt0 + m) * 16 + ln][koff]);

#pragma unroll 2
    for (int k = 0; k < K; k += 4) {
        const int kn = (k + 4 < K) ? (k + 4) : k;   // uniform; last prefetch dummy
        v2f an[MT], bn;
        bn.x = LW[kn + koff][ncol];
        bn.y = LW[kn + koff + 1][ncol];
#pragma unroll
        for (int m = 0; m < MT; ++m)
            an[m] = *(const v2f*)(&LX[(mt0 + m) * 16 + ln][kn + koff]);
#pragma unroll
        for (int m = 0; m < MT; ++m)
            acc[m] = __builtin_amdgcn_wmma_f32_16x16x4_f32(
                false, a[m], false, b, (short)0, acc[m], false, false);
        b = bn;
#pragma unroll
        for (int m = 0; m < MT; ++m) a[m] = an[m];
    }

    // Store D: VGPR r -> row r (+8 for upper lane half), col = ncol.
#pragma unroll
    for (int m = 0; m < MT; ++m) {
        int base = row0 + (mt0 + m) * 16 + ((lane < 16) ? 0 : 8);
#pragma unroll
        for (int r = 0; r < 8; ++r) {
            int gr = base + r;
            if (gr < nrows) Y[(size_t)gr * NCOLS + ncol] = acc[m][r];
        }
    }
}

// ---------------------------------------------------------------------------
// Launch
// ---------------------------------------------------------------------------
extern "C" void kernel_launch(void* const* d_in, const int* in_sizes, int n_in,
                              void* d_out, int out_size, void* d_ws, size_t ws_size,
                              hipStream_t stream) {
    const float* x  = (const float*)d_in[0];
    const int*   ei = (const int*)d_in[1];
    const float* W1 = (const float*)d_in[2];
    const float* b1 = (const float*)d_in[3];
    const float* W2 = (const float*)d_in[4];
    const float* b2 = (const float*)d_in[5];

    const int DIN = 128, DHID = 128, DOUT = 64;
    const int N = in_sizes[0] / DIN;
    const int E = in_sizes[1] / 2;
    const int* src = ei;
    const int* dst = ei + E;
    float* out = (float*)d_out;

    // Workspace carve-out (~110 MB): deg | norm | bufA | bufB
    char* ws = (char*)d_ws;
    size_t off = 0;
    auto carve = [&](size_t bytes) -> float* {
        float* p = (float*)(ws + off);
        off += (bytes + 255) & ~(size_t)255;
        return p;
    };
    float* deg  = carve((size_t)N * sizeof(float));          // becomes deg^{-1/2}
    float* norm = carve((size_t)E * sizeof(float));
    float* bufA = carve((size_t)N * DHID * sizeof(float));   // h1, then h2
    float* bufB = carve((size_t)N * DHID * sizeof(float));   // agg1 / relu in place

    const int TB = 256;
    const int gN = (N + TB - 1) / TB;
    const int gE = (E + TB - 1) / TB;

    // ---- normalization coefficients ------------------------------------
    fill_zero_kernel<<<gN, TB, 0, stream>>>(deg, N);
    degree_kernel<<<gE, TB, 0, stream>>>(dst, deg, E);
    inv_sqrt_kernel<<<gN, TB, 0, stream>>>(deg, N);
    norm_kernel<<<gE, TB, 0, stream>>>(src, dst, deg, norm, E);

    // ---- layer 1: h1 = relu( scatter(gather(x@W1)*norm) + b1 ) ---------
    gemm_wmma_kernel<DHID><<<(N + 127) / 128, TB, 0, stream>>>(x, W1, bufA, N);

    long long t1 = (long long)N * DHID;
    bias_fill_kernel<<<(int)((t1 + TB - 1) / TB), TB, 0, stream>>>(bufB, b1, DHID, t1);
    long long a1 = (long long)E * (DHID / 4);
    aggregate_kernel<DHID><<<(int)((a1 + TB - 1) / TB), TB, 0, stream>>>(
        bufA, norm, src, dst, bufB, E);
    relu_kernel<<<(int)((t1 + TB - 1) / TB), TB, 0, stream>>>(bufB, t1);

    // ---- layer 2: out = scatter(gather(h1@W2)*norm) + b2 ---------------
    gemm_wmma_kernel<DOUT><<<(N + 127) / 128, TB, 0, stream>>>(bufB, W2, bufA, N);

    long long t2 = (long long)N * DOUT;
    bias_fill_kernel<<<(int)((t2 + TB - 1) / TB), TB, 0, stream>>>(out, b2, DOUT, t2);
    long long a2 = (long long)E * (DOUT / 4);
    aggregate_kernel<DOUT><<<(int)((a2 + TB - 1) / TB), TB, 0, stream>>>(
        bufA, norm, src, dst, out, E);
}